// Embedding_824633721067
// MI455X (gfx1250) — compile-verified
//
#include <hip/hip_runtime.h>

#define NPTS   100000
#define CIN    6
#define COUT   48
#define KOFF   124
#define PAD_P  12000
#define BN_EPS 1e-5f

typedef float v8f __attribute__((ext_vector_type(8)));
typedef float v2f __attribute__((ext_vector_type(2)));

// ---------------------------------------------------------------------------
// Kernel 1: dense center GEMM  out[n, :] = feats[n, :] @ W_center
// One wave = one 16-row tile; 3 column tiles of 16; K padded 6 -> 8.
// V_WMMA_F32_16X16X4_F32: A 16x4 (v2f/lane), B 4x16 (v2f/lane), C/D v8f.
// All loads are unconditional (always in-bounds); zero-padding of the K=6,7
// slots is done with selects (v_cndmask), not EXEC-masked loads.
// ---------------------------------------------------------------------------
__global__ void __launch_bounds__(256) center_gemm(const float* __restrict__ feats,
                                                   const float* __restrict__ Wc,
                                                   float* __restrict__ out) {
    const int wave = (blockIdx.x * blockDim.x + threadIdx.x) >> 5;
    if (wave >= NPTS / 16) return;                 // whole-wave uniform exit
    const int lane = threadIdx.x & 31;
    const int m    = lane & 15;
    const int hi   = lane >> 4;
    const int rowBase = wave * 16;
    const float* fr = feats + (size_t)(rowBase + m) * CIN;

    // A: VGPR0 = K0 (lanes<16) / K2 (lanes>=16); VGPR1 = K1 / K3. 8B-aligned b64 loads.
    v2f a0 = *(const v2f*)(fr + (hi ? 2 : 0));     // K-step 0: K=0..3
    v2f t45 = *(const v2f*)(fr + 4);               // K=4,5 (always in-bounds)
    v2f a1;
    a1.x = hi ? 0.f : t45.x;                       // K-step 1: K=4,5; K=6,7 pad
    a1.y = hi ? 0.f : t45.y;

    const int r0 = (hi ? 2 : 0) * COUT;
    const int r1 = (hi ? 3 : 1) * COUT;

    v8f acc[3];
    for (int tI = 0; tI < 3; ++tI)
        for (int j = 0; j < 8; ++j) acc[tI][j] = 0.f;

    for (int tI = 0; tI < 3; ++tI) {
        const int col = tI * 16 + m;
        v2f b0, b1;
        b0.x = Wc[r0 + col];
        b0.y = Wc[r1 + col];
        const float w4 = Wc[4 * COUT + col];
        const float w5 = Wc[5 * COUT + col];
        b1.x = hi ? 0.f : w4;
        b1.y = hi ? 0.f : w5;
        acc[tI] = __builtin_amdgcn_wmma_f32_16x16x4_f32(false, a0, false, b0,
                                                        (short)0, acc[tI], false, false);
        acc[tI] = __builtin_amdgcn_wmma_f32_16x16x4_f32(false, a1, false, b1,
                                                        (short)0, acc[tI], false, false);
    }

    // C/D layout: VGPR j: lanes<16 -> (M=j, N=m); lanes>=16 -> (M=j+8, N=m)
    for (int tI = 0; tI < 3; ++tI)
        for (int j = 0; j < 8; ++j)
            out[(size_t)(rowBase + j + (hi ? 8 : 0)) * COUT + tI * 16 + m] = acc[tI][j];
}

// ---------------------------------------------------------------------------
// Kernel 2: rulebook gather-GEMM-scatter.
// Padded slots have in_idx == out_idx == 0; valid pairs never have in == out
// (offset 0 excluded), so validity = (in != out) -- the bool mask input is
// not needed. Fully-padded tiles (wave-uniform) exit before any work.
// Invalid rows get zero A-rows (select, not branch) and skip their atomics.
// ---------------------------------------------------------------------------
__global__ void __launch_bounds__(256) scatter_gemm(const float* __restrict__ feats,
                                                    const float* __restrict__ Woff,
                                                    const int*   __restrict__ in_idx,
                                                    const int*   __restrict__ out_idx,
                                                    float*       __restrict__ out) {
    const int WPK  = PAD_P / 16;                   // 750 waves per offset
    const int wave = (blockIdx.x * blockDim.x + threadIdx.x) >> 5;
    if (wave >= KOFF * WPK) return;
    const int k  = wave / WPK;
    const int p0 = (wave % WPK) * 16;
    const int lane = threadIdx.x & 31;
    const int m    = lane & 15;
    const int hi   = lane >> 4;

    const int  p  = p0 + m;
    const int  ri = in_idx [k * PAD_P + p];
    const int  ro = out_idx[k * PAD_P + p];
    const bool valid = (ri != ro);

    // Rulebook padding is a suffix: whole-tile-invalid waves are common.
    if ((unsigned)__ballot(valid) == 0u) return;   // wave-uniform

    const float* fr = feats + (size_t)ri * CIN;    // ri = 0 for padding: in-bounds
    const float* Wk = Woff + (size_t)k * CIN * COUT;

    v2f ta0 = *(const v2f*)(fr + (hi ? 2 : 0));
    v2f t45 = *(const v2f*)(fr + 4);
    v2f a0, a1;
    a0.x = valid ? ta0.x : 0.f;
    a0.y = valid ? ta0.y : 0.f;
    a1.x = (valid && !hi) ? t45.x : 0.f;
    a1.y = (valid && !hi) ? t45.y : 0.f;

    const int r0 = (hi ? 2 : 0) * COUT;
    const int r1 = (hi ? 3 : 1) * COUT;

    v8f acc[3];
    for (int tI = 0; tI < 3; ++tI)
        for (int j = 0; j < 8; ++j) acc[tI][j] = 0.f;

    for (int tI = 0; tI < 3; ++tI) {
        const int col = tI * 16 + m;
        v2f b0, b1;
        b0.x = Wk[r0 + col];
        b0.y = Wk[r1 + col];
        const float w4 = Wk[4 * COUT + col];
        const float w5 = Wk[5 * COUT + col];
        b1.x = hi ? 0.f : w4;
        b1.y = hi ? 0.f : w5;
        acc[tI] = __builtin_amdgcn_wmma_f32_16x16x4_f32(false, a0, false, b0,
                                                        (short)0, acc[tI], false, false);
        acc[tI] = __builtin_amdgcn_wmma_f32_16x16x4_f32(false, a1, false, b1,
                                                        (short)0, acc[tI], false, false);
    }

    // Broadcast each tile-row's destination (-1 = invalid) across the wave.
    const int sro = valid ? ro : -1;
    int rdest[8];
    for (int j = 0; j < 8; ++j)
        rdest[j] = __shfl(sro, j + (hi ? 8 : 0), 32);

    for (int tI = 0; tI < 3; ++tI)
        for (int j = 0; j < 8; ++j)
            if (rdest[j] >= 0)
                unsafeAtomicAdd(out + (size_t)rdest[j] * COUT + tI * 16 + m, acc[tI][j]);
}

// ---------------------------------------------------------------------------
// Kernel 3: per-channel sum / sum-of-squares (ws[c], ws[48+c]).
// ---------------------------------------------------------------------------
__global__ void __launch_bounds__(256) bn_reduce(const float* __restrict__ out,
                                                 float* __restrict__ sums) {
    __shared__ float s1[256];
    __shared__ float s2[256];
    const int c = blockIdx.x;
    float a = 0.f, b = 0.f;
    for (int n = blockIdx.y * 256 + threadIdx.x; n < NPTS; n += gridDim.y * 256) {
        float v = out[(size_t)n * COUT + c];
        a += v; b += v * v;
    }
    s1[threadIdx.x] = a; s2[threadIdx.x] = b;
    __syncthreads();
    for (int off = 128; off > 0; off >>= 1) {
        if (threadIdx.x < off) {
            s1[threadIdx.x] += s1[threadIdx.x + off];
            s2[threadIdx.x] += s2[threadIdx.x + off];
        }
        __syncthreads();
    }
    if (threadIdx.x == 0) {
        unsafeAtomicAdd(&sums[c],        s1[0]);
        unsafeAtomicAdd(&sums[COUT + c], s2[0]);
    }
}

// ---------------------------------------------------------------------------
// Kernel 4: in-place normalize + affine + ReLU.
// ---------------------------------------------------------------------------
__global__ void __launch_bounds__(256) bn_apply(float* __restrict__ out,
                                                const float* __restrict__ sums,
                                                const float* __restrict__ gamma,
                                                const float* __restrict__ beta) {
    const int idx = blockIdx.x * 256 + threadIdx.x;
    if (idx >= NPTS * COUT) return;
    const int c = idx % COUT;
    const float inv_n = 1.f / (float)NPTS;
    const float mean  = sums[c] * inv_n;
    const float var   = sums[COUT + c] * inv_n - mean * mean;
    const float y = (out[idx] - mean) * rsqrtf(var + BN_EPS) * gamma[c] + beta[c];
    out[idx] = y > 0.f ? y : 0.f;
}

__global__ void zero_ws(float* __restrict__ p) {
    if (threadIdx.x < 2 * COUT) p[threadIdx.x] = 0.f;
}

// ---------------------------------------------------------------------------
extern "C" void kernel_launch(void* const* d_in, const int* in_sizes, int n_in,
                              void* d_out, int out_size, void* d_ws, size_t ws_size,
                              hipStream_t stream) {
    (void)in_sizes; (void)n_in; (void)out_size; (void)ws_size;
    const float* feats   = (const float*)d_in[0];
    const float* Wc      = (const float*)d_in[1];
    const float* Woff    = (const float*)d_in[2];
    const float* gamma   = (const float*)d_in[3];
    const float* beta    = (const float*)d_in[4];
    const int*   in_idx  = (const int*)d_in[5];
    const int*   out_idx = (const int*)d_in[6];
    // d_in[7] (mask) intentionally unused: padding detected via in_idx==out_idx.
    float* out  = (float*)d_out;
    float* sums = (float*)d_ws;   // 96 floats

    zero_ws<<<1, 128, 0, stream>>>(sums);

    {   // center GEMM: 6250 waves (exactly covers 100000 rows)
        const int waves  = NPTS / 16;
        const int blocks = (waves * 32 + 255) / 256;
        center_gemm<<<blocks, 256, 0, stream>>>(feats, Wc, out);
    }
    {   // scatter GEMM: 124 * 750 = 93000 waves
        const int waves  = KOFF * (PAD_P / 16);
        const int blocks = (waves * 32) / 256;
        scatter_gemm<<<blocks, 256, 0, stream>>>(feats, Woff, in_idx, out_idx, out);
    }
    bn_reduce<<<dim3(COUT, 64), 256, 0, stream>>>(out, sums);
    bn_apply<<<(NPTS * COUT) / 256, 256, 0, stream>>>(out, sums, gamma, beta);
}